// ImagePool_30305289240936
// MI455X (gfx1250) — compile-verified
//
#include <hip/hip_runtime.h>
#include <stdint.h>

// Problem geometry (from reference): N=128, H=256, W=256, C=3, CAP=50
#define IMG_N 128
#define IMG_ELEMS (256 * 256 * 3)          // 196608 floats per image
#define VECS_PER_IMG (IMG_ELEMS / 4)       // 49152 float4 per image
#define TPB 256
#define STAGES 4                            // 4 x b128 per thread per block
#define BLOCKS_PER_IMG (VECS_PER_IMG / (TPB * STAGES))  // 48

// ---------------------------------------------------------------------------
// Kernel 1: resolve the sequential pool-mutation chain into a per-image
// source descriptor.  src[i] >= 0  -> copy inputs[src[i]]
//                     src[i] <  0  -> copy pool[-src[i]-1]
// O(N^2) = 16K integer ops total; runs in one tiny block.
// ---------------------------------------------------------------------------
__global__ void pool_map_kernel(const int* __restrict__ swap_mask,
                                const int* __restrict__ idx,
                                int* __restrict__ src) {
    int i = threadIdx.x;
    if (i >= IMG_N) return;
    int s = i;  // default: pass-through of inputs[i]
    if (swap_mask[i] != 0) {
        int j = idx[i];
        s = -(j + 1);  // original pool slot j, unless overwritten earlier
        for (int p = i - 1; p >= 0; --p) {
            if (swap_mask[p] != 0 && idx[p] == j) { s = p; break; }
        }
    }
    src[i] = s;
}

// ---------------------------------------------------------------------------
// Kernel 2: streaming gather-copy through the gfx1250 async Global<->LDS DMA
// path.  Each thread moves STAGES x 16B: batch async loads into a private LDS
// slot, one s_wait_asynccnt, batch async stores out.  512B per instruction
// per wave; ~49K waves in flight to hide the single latency stall.
// ---------------------------------------------------------------------------
__global__ void __launch_bounds__(TPB)
pool_copy_kernel(const float* __restrict__ inputs,
                 const float* __restrict__ pool,
                 const int* __restrict__ src,
                 float* __restrict__ out) {
    __shared__ float4 smem[TPB * STAGES];   // 16 KB per block

    const int img = blockIdx.y;
    const int s = src[img];                 // uniform per block
    const float* sbase = (s >= 0) ? (inputs + (size_t)s * IMG_ELEMS)
                                  : (pool + (size_t)(-s - 1) * IMG_ELEMS);
    const float4* gsrc = (const float4*)sbase;
    float4* gdst = (float4*)(out + (size_t)img * IMG_ELEMS);

    const size_t v0 = (size_t)blockIdx.x * (TPB * STAGES) + threadIdx.x;

    // Wave-relative LDS byte offset = low 32 bits of the generic pointer
    // (LDS aperture places the offset in addr[31:0]).
    const uint32_t lds0 = (uint32_t)(uintptr_t)(&smem[threadIdx.x]);

    // Issue STAGES async b128 loads (ASYNCcnt += STAGES).
#pragma unroll
    for (int k = 0; k < STAGES; ++k) {
        uint64_t ga = (uint64_t)(uintptr_t)(gsrc + v0 + (size_t)k * TPB);
        uint32_t la = lds0 + (uint32_t)(k * TPB * sizeof(float4));
        asm volatile("global_load_async_to_lds_b128 %0, %1, off"
                     :: "v"(la), "v"(ga)
                     : "memory");
    }
    // Loads complete in order; wait for all before reading LDS back out.
    asm volatile("s_wait_asynccnt 0" ::: "memory");

    // Issue STAGES async b128 stores (LDS -> global).
#pragma unroll
    for (int k = 0; k < STAGES; ++k) {
        uint64_t ga = (uint64_t)(uintptr_t)(gdst + v0 + (size_t)k * TPB);
        uint32_t la = lds0 + (uint32_t)(k * TPB * sizeof(float4));
        asm volatile("global_store_async_from_lds_b128 %0, %1, off"
                     :: "v"(ga), "v"(la)
                     : "memory");
    }
    // Drain before wave end (s_endpgm's implicit wait would also cover this).
    asm volatile("s_wait_asynccnt 0" ::: "memory");
}

// ---------------------------------------------------------------------------
// Host-side launch.  d_in order per setup_inputs(): inputs, pool, swap_mask,
// idx.  d_ws holds the 128-entry source map (512 B).
// ---------------------------------------------------------------------------
extern "C" void kernel_launch(void* const* d_in, const int* in_sizes, int n_in,
                              void* d_out, int out_size, void* d_ws, size_t ws_size,
                              hipStream_t stream) {
    const float* inputs    = (const float*)d_in[0];
    const float* pool      = (const float*)d_in[1];
    const int*   swap_mask = (const int*)d_in[2];
    const int*   idx       = (const int*)d_in[3];
    float*       out       = (float*)d_out;
    int*         src       = (int*)d_ws;

    pool_map_kernel<<<1, IMG_N, 0, stream>>>(swap_mask, idx, src);

    dim3 grid(BLOCKS_PER_IMG, IMG_N);
    pool_copy_kernel<<<grid, TPB, 0, stream>>>(inputs, pool, src, out);
}